// ChunkedGeoSparseLinear_27762668601602
// MI455X (gfx1250) — compile-verified
//
#include <hip/hip_runtime.h>

// Problem constants (from reference)
#define IN_F    4096
#define OUT_F   4096
#define KNZ     16

// Tiling
#define ROWS    16          // x rows staged per workgroup
#define RSTRIDE 4100        // LDS row stride in dwords: ≡4 (mod 64) -> bank-conflict-free
                            // across the 16 row-lanes, and rows stay 16B aligned for B128
#define TSTRIDE 65          // transpose tile stride (odd -> conflict-free)
#define BLOCK   1024        // 32 waves
#define O_ROUND 64          // outputs per round = 32 waves * 2
#define ROUNDS  (OUT_F / O_ROUND)
#define LDS_DWORDS (ROWS * RSTRIDE + ROWS * TSTRIDE)   // 65600 + 1040 = 66640 dwords (~260 KB + 4 KB)

typedef int v4i __attribute__((ext_vector_type(4)));
typedef __attribute__((address_space(1))) v4i* global_v4i_ptr;
typedef __attribute__((address_space(3))) v4i* lds_v4i_ptr;

// ---------------- CDNA5 async global->LDS copy helpers ----------------
__device__ __forceinline__ void async_b128(const float* g, float* l) {
#if __has_builtin(__builtin_amdgcn_global_load_async_to_lds_b128)
    __builtin_amdgcn_global_load_async_to_lds_b128(
        (global_v4i_ptr)(v4i*)g,
        (lds_v4i_ptr)(v4i*)l,
        0, 0);
#else
    unsigned laddr = (unsigned)(unsigned long long)
        (__attribute__((address_space(3))) float*)l;
    asm volatile("global_load_async_to_lds_b128 %0, %1, off"
                 :: "v"(laddr), "v"(g) : "memory");
#endif
}

__device__ __forceinline__ void wait_async0() {
#if __has_builtin(__builtin_amdgcn_s_wait_asynccnt)
    __builtin_amdgcn_s_wait_asynccnt(0);
#else
    asm volatile("s_wait_asynccnt 0x0" ::: "memory");
#endif
}

// ---------------- Pre-pass: int64 indices -> int32 ----------------
__global__ void pack_idx_kernel(const long long* __restrict__ idx64,
                                int* __restrict__ idx32, int n) {
    int t = blockIdx.x * blockDim.x + threadIdx.x;
    if (t < n) idx32[t] = (int)idx64[t];
}

// ---------------- Main kernel ----------------
extern __shared__ float smem[];

__global__ void __launch_bounds__(BLOCK, 1)
gsl_main_kernel(const float* __restrict__ x,
                const int*   __restrict__ idx32,
                const float* __restrict__ w,
                const float* __restrict__ bias,
                float*       __restrict__ out) {
    float* xs   = smem;                       // ROWS x RSTRIDE staged x rows
    float* tile = smem + ROWS * RSTRIDE;      // ROWS x O_ROUND transpose tile (stride TSTRIDE)

    const int t  = threadIdx.x;
    const int n0 = blockIdx.x * ROWS;

    // ---- Phase 1: async-stage 16 fp32 rows of x into LDS (256 KB, B128 chunks) ----
    // chunks per row = IN_F/4 = 1024; total 16384 chunks; 16 per thread.
    for (int i = t; i < ROWS * (IN_F / 4); i += BLOCK) {
        const int row = i >> 10;
        const int c4  = i & 1023;
        async_b128(x + (size_t)(n0 + row) * IN_F + c4 * 4,
                   xs + row * RSTRIDE + c4 * 4);
    }
    wait_async0();
    __syncthreads();

    // ---- Phase 2: gather-dot-accumulate ----
    // Lane mapping: lane = (h<<4)|r ; r = staged row (0..15), h selects which of the
    // wave's two outputs. Gather address = r*RSTRIDE + idx  -> banks r*4+idx mod 64,
    // distinct across the 16 lanes of a half (RSTRIDE % 64 == 4).
    const int lane  = t & 31;
    const int r     = lane & 15;
    const int h     = lane >> 4;
    const int wv    = t >> 5;            // wave id 0..31
    const int ocol  = wv * 2 + h;        // 0..63 within the round
    const float* xrow = xs + r * RSTRIDE;

    // store-phase mapping (coalesced): thread t -> (row = t/64, col = t%64)
    const int str = t >> 6;
    const int stc = t & 63;

#pragma unroll 1
    for (int ro = 0; ro < ROUNDS; ++ro) {
        const int o = ro * O_ROUND + ocol;
        const int4*   ip = (const int4*)(idx32 + o * KNZ);
        const float4* wp = (const float4*)(w + o * KNZ);

        float acc = 0.0f;
#pragma unroll
        for (int q = 0; q < 4; ++q) {
            const int4   iv = ip[q];
            const float4 wf = wp[q];
            acc = fmaf(xrow[iv.x], wf.x, acc);
            acc = fmaf(xrow[iv.y], wf.y, acc);
            acc = fmaf(xrow[iv.z], wf.z, acc);
            acc = fmaf(xrow[iv.w], wf.w, acc);
        }

        // transpose through LDS so global stores are contiguous 128B lines
        tile[r * TSTRIDE + ocol] = acc;
        __syncthreads();
        out[(size_t)(n0 + str) * OUT_F + ro * O_ROUND + stc] =
            tile[str * TSTRIDE + stc] + bias[ro * O_ROUND + stc];
        __syncthreads();
    }
}

// ---------------- Host launch ----------------
extern "C" void kernel_launch(void* const* d_in, const int* in_sizes, int n_in,
                              void* d_out, int out_size, void* d_ws, size_t ws_size,
                              hipStream_t stream) {
    const float*     x     = (const float*)d_in[0];
    const long long* idx64 = (const long long*)d_in[1];
    const float*     w     = (const float*)d_in[2];
    const float*     bias  = (const float*)d_in[3];
    float*           out   = (float*)d_out;

    int* idx32 = (int*)d_ws;  // 65536 * 4B = 256 KB scratch

    const int nidx = OUT_F * KNZ;
    pack_idx_kernel<<<(nidx + 255) / 256, 256, 0, stream>>>(idx64, idx32, nidx);

    const int N = in_sizes[0] / IN_F;   // 4096
    const int lds_bytes = LDS_DWORDS * 4;
    (void)hipFuncSetAttribute((const void*)gsl_main_kernel,
                              hipFuncAttributeMaxDynamicSharedMemorySize, lds_bytes);

    gsl_main_kernel<<<N / ROWS, BLOCK, lds_bytes, stream>>>(x, idx32, w, bias, out);
}